// MaxKGIN_51161650430041
// MI455X (gfx1250) — compile-verified
//
#include <hip/hip_runtime.h>
#include <hip/hip_bf16.h>

typedef __attribute__((ext_vector_type(2))) float v2f;
typedef __attribute__((ext_vector_type(8))) float v8f;

#define IN_SIZE 128
#define HID     96
#define OUT_F   40
#define NLAYERS 3
#define TOPK    32
#define MSUB    4          // 16-row m-subtiles per block (64 rows/block)
#define KMAX    128
#define KPAD_MAX (KMAX + 2)

// ---------------------------------------------------------------------------
// f32 WMMA GEMM: C[M,N] = A[M,K] @ B[K,N] + bias[N], optional ReLU.
// Block = 128 threads (4 waves). Block processes MSUB 16-row m-subtiles,
// keeping zero-padded B resident in LDS (staged once).
// LDS layouts padded to Kpad = K+2 words per row so the 16-lane fragment
// reads are bank-conflict-free and 8-byte aligned:
//   As [16][Kpad]        row-major A subtile
//   BsT[Npad][Kpad]      B transposed (col-major) -> fragment = one b64 load
// Each wave owns 16x16 output tiles (n-tiles strided by 4), accumulating
// over K in steps of 4 with V_WMMA_F32_16X16X4_F32.
// Requires K % 4 == 0, K <= 128, N <= 96.
// ---------------------------------------------------------------------------
__global__ void maxkgin_wmma_gemm(const float* __restrict__ A,
                                  const float* __restrict__ B,
                                  const float* __restrict__ bias,
                                  float* __restrict__ C,
                                  int M, int K, int N, int relu) {
    __shared__ float As[16 * KPAD_MAX];        // 8.1 KB
    __shared__ float BsT[96 * KPAD_MAX];       // 48.8 KB

    const int tid  = threadIdx.x;
    const int NT   = (N + 15) / 16;
    const int Npad = NT * 16;
    const int Kpad = K + 2;

    // Stage B transposed: global read coalesced over columns, LDS write
    // stride Kpad words (conflict-free permutation for K=96/128).
    for (int idx = tid; idx < K * Npad; idx += 128) {
        int k = idx / Npad, col = idx - k * Npad;
        BsT[col * Kpad + k] = (col < N) ? B[(size_t)k * N + col] : 0.0f;
    }

    const int wave = tid >> 5;
    const int lane = tid & 31;
    const int half = lane >> 4;   // 0: lanes 0-15, 1: lanes 16-31
    const int l16  = lane & 15;
    const int aoff = l16 * Kpad + 2 * half;    // A fragment base (words)

    for (int s = 0; s < MSUB; ++s) {
        const int row0 = (blockIdx.x * MSUB + s) * 16;
        if (s > 0) __syncthreads();            // previous subtile compute done
        for (int idx = tid; idx < 16 * K; idx += 128) {
            int r = idx / K, c = idx - r * K;
            As[r * Kpad + c] = (row0 + r < M) ? A[(size_t)(row0 + r) * K + c]
                                              : 0.0f;
        }
        __syncthreads();                       // As (and BsT on s==0) visible

        for (int nt = wave; nt < NT; nt += 4) {
            const int n0   = nt * 16;
            const int boff = (n0 + l16) * Kpad + 2 * half;
            v8f acc = {0.f, 0.f, 0.f, 0.f, 0.f, 0.f, 0.f, 0.f};

            for (int k0 = 0; k0 < K; k0 += 4) {
                // A frag: lanes<16 hold A[m, k0..k0+1]; lanes>=16 A[m, k0+2..3]
                const v2f a = *reinterpret_cast<const v2f*>(&As[aoff + k0]);
                // B frag: lanes<16 hold B[k0..k0+1, col]; lanes>=16 B[k0+2..3, col]
                const v2f b = *reinterpret_cast<const v2f*>(&BsT[boff + k0]);
                acc = __builtin_amdgcn_wmma_f32_16x16x4_f32(
                    false, a, false, b, (short)0, acc, false, false);
            }

            // D element v_r is row (row0 + r + 8*half), col = n0 + l16.
            const int col = n0 + l16;
            if (col < N) {
                const float bv = bias[col];
                const int rbase = row0 + 8 * half;
#pragma unroll
                for (int r = 0; r < 8; ++r) {
                    float o = acc[r] + bv;
                    if (relu) o = fmaxf(o, 0.0f);
                    const int row = rbase + r;
                    if (row < M) C[(size_t)row * N + col] = o;
                }
            }
        }
    }
}

// ---------------------------------------------------------------------------
// Exact top-K per row (keep 32 of 96): wave per row, 3 values per lane.
// Radix-select the 32nd-largest monotone uint key, zero everything below.
// ---------------------------------------------------------------------------
__global__ void maxkgin_maxk(float* __restrict__ h, int n_rows) {
    const int gid  = blockIdx.x * blockDim.x + threadIdx.x;
    const int row  = gid >> 5;
    const int lane = gid & 31;
    if (row >= n_rows) return;

    float* rp = h + (size_t)row * HID;
    unsigned key[3];
#pragma unroll
    for (int j = 0; j < 3; ++j) {
        unsigned u = __float_as_uint(rp[lane + 32 * j]);
        key[j] = u ^ ((u & 0x80000000u) ? 0xFFFFFFFFu : 0x80000000u);
    }

    unsigned prefix = 0;
    int remaining = TOPK;
#pragma unroll
    for (int bit = 31; bit >= 0; --bit) {
        const unsigned cb = 1u << bit;
        const unsigned hi = ~((cb << 1) - 1u);   // bits above `bit` (0 at bit 31)
        int c = 0;
#pragma unroll
        for (int j = 0; j < 3; ++j)
            c += (((key[j] & hi) == prefix) && (key[j] & cb)) ? 1 : 0;
        for (int off = 16; off > 0; off >>= 1)
            c += __shfl_xor(c, off, 32);
        if (c >= remaining) prefix |= cb;
        else                remaining -= c;
    }
#pragma unroll
    for (int j = 0; j < 3; ++j)
        if (key[j] < prefix) rp[lane + 32 * j] = 0.0f;
}

__global__ void maxkgin_zero(float* __restrict__ p, int n) {
    int i = blockIdx.x * blockDim.x + threadIdx.x;
    if (i < n) p[i] = 0.0f;
}

// ---------------------------------------------------------------------------
// GIN sum aggregation: neigh[dst[e]] += h[src[e]].  Wave per edge,
// 3 floats per lane; skip zeros produced by maxk (2/3 of entries).
// ---------------------------------------------------------------------------
__global__ void maxkgin_edge_scatter(const float* __restrict__ h,
                                     const int* __restrict__ src,
                                     const int* __restrict__ dst,
                                     float* __restrict__ neigh,
                                     int n_edges) {
    const int gid  = blockIdx.x * blockDim.x + threadIdx.x;
    const int e    = gid >> 5;
    const int lane = gid & 31;
    if (e >= n_edges) return;
    const int s = src[e], d = dst[e];
    const float* hs = h + (size_t)s * HID;
    float* nd = neigh + (size_t)d * HID;
#pragma unroll
    for (int j = 0; j < 3; ++j) {
        float val = hs[lane + 32 * j];
        if (val != 0.0f) atomicAdd(nd + lane + 32 * j, val);
    }
}

__global__ void maxkgin_combine(const float* __restrict__ hB,
                                const float* __restrict__ nB,
                                const float* __restrict__ eps, int layer,
                                float* __restrict__ hA, int n) {
    int i = blockIdx.x * blockDim.x + threadIdx.x;
    if (i < n) hA[i] = (1.0f + eps[layer]) * hB[i] + nB[i];
}

extern "C" void kernel_launch(void* const* d_in, const int* in_sizes, int n_in,
                              void* d_out, int out_size, void* d_ws, size_t ws_size,
                              hipStream_t stream) {
    (void)n_in; (void)out_size; (void)ws_size;

    const float* x     = (const float*)d_in[0];
    const int*   src   = (const int*)  d_in[1];
    const int*   dst   = (const int*)  d_in[2];
    const float* W_in  = (const float*)d_in[3];
    const float* b_in  = (const float*)d_in[4];
    const float* Wl    = (const float*)d_in[5];
    const float* bl    = (const float*)d_in[6];
    const float* eps   = (const float*)d_in[7];
    const float* W_out = (const float*)d_in[8];
    const float* b_out = (const float*)d_in[9];
    float* out = (float*)d_out;

    const int M       = in_sizes[0] / IN_SIZE;   // 50000
    const int n_edges = in_sizes[1];             // 800000

    float* hA = (float*)d_ws;
    float* hB = hA + (size_t)M * HID;
    float* nB = hB + (size_t)M * HID;

    const int rows_per_blk = 16 * MSUB;
    const int gemm_grid    = (M + rows_per_blk - 1) / rows_per_blk;
    const int elems        = M * HID;

    // h = relu(x @ W_in + b_in)
    maxkgin_wmma_gemm<<<gemm_grid, 128, 0, stream>>>(x, W_in, b_in, hA,
                                                     M, IN_SIZE, HID, 1);

    for (int l = 0; l < NLAYERS; ++l) {
        // hB = hA @ Wl[l] + bl[l]
        maxkgin_wmma_gemm<<<gemm_grid, 128, 0, stream>>>(
            hA, Wl + (size_t)l * HID * HID, bl + (size_t)l * HID, hB,
            M, HID, HID, 0);
        // hB = maxk(hB, 32)
        maxkgin_maxk<<<(M * 32 + 255) / 256, 256, 0, stream>>>(hB, M);
        // neigh = segment_sum(hB[src], dst)
        maxkgin_zero<<<(elems + 255) / 256, 256, 0, stream>>>(nB, elems);
        maxkgin_edge_scatter<<<(n_edges * 32 + 255) / 256, 256, 0, stream>>>(
            hB, src, dst, nB, n_edges);
        // hA = (1+eps[l])*hB + neigh
        maxkgin_combine<<<(elems + 255) / 256, 256, 0, stream>>>(
            hB, nB, eps, l, hA, elems);
    }

    // out = hA @ W_out + b_out
    maxkgin_wmma_gemm<<<gemm_grid, 128, 0, stream>>>(hA, W_out, b_out, out,
                                                     M, HID, OUT_F, 0);
}